// FactorizedVariationalPolicy_18622978195982
// MI455X (gfx1250) — compile-verified
//
#include <hip/hip_runtime.h>
#include <hip/hip_bf16.h>
#include <math.h>

typedef __attribute__((ext_vector_type(2))) float v2f;
typedef __attribute__((ext_vector_type(8))) float v8f;

// D = A(16x4,f32) * B(4x16,f32) + C(16x16,f32)  -- exact fp32 accumulation
#define WMMA_F32(A, B, C) \
  __builtin_amdgcn_wmma_f32_16x16x4_f32(false, (A), false, (B), (short)0, (C), false, false)

__device__ __forceinline__ v2f mkv2(float a, float b) { v2f r; r.x = a; r.y = b; return r; }

// Wave-local LDS ordering: each wave owns a private LDS region; DS ops from one
// wave complete in order, so s_wait_dscnt 0 (plus compiler memory barrier) is a
// sufficient fence without a workgroup barrier.
__device__ __forceinline__ void wave_lds_fence() {
  asm volatile("s_wait_dscnt 0" ::: "memory");
}

constexpr int Bsz = 32768;
constexpr int D   = 64;
constexpr int H   = 32;
constexpr int TILES = 4;                              // 16-row tiles per wave
constexpr int WAVES = 8;                              // waves per block (256 thr)
constexpr int ROWS_PER_BLOCK = WAVES * TILES * 16;    // 512

// ---------------------------------------------------------------------------
// Kernel 1: all D per-feature encoders. One block = (one feature d, 512 rows).
// h1 built in-register in WMMA A-layout; h2 = relu(h1@W2+b2) via f32 WMMA;
// mu/logvar via one WMMA chain with {Wmu,Wlv} packed in B columns 0/1.
// Writes mu, logvar to d_out and z = mu + eps*exp(0.5*lv) to workspace.
// ---------------------------------------------------------------------------
__global__ __launch_bounds__(256) void encoder_kernel(
    const float* __restrict__ x,   const float* __restrict__ eps,
    const float* __restrict__ W1,  const float* __restrict__ b1,
    const float* __restrict__ W2,  const float* __restrict__ b2,
    const float* __restrict__ Wmu, const float* __restrict__ bmu,
    const float* __restrict__ Wlv, const float* __restrict__ blv,
    float* __restrict__ mu_out, float* __restrict__ lv_out,
    float* __restrict__ z_ws)
{
  __shared__ float lds[WAVES * 16 * H];
  const int lane = threadIdx.x & 31;
  const int wave = threadIdx.x >> 5;
  const int col  = lane & 15;   // N (B/C cols) or M (A rows)
  const int hi   = lane >> 4;   // half-wave: K-offset selector
  const int d    = blockIdx.y;
  float* wl = &lds[wave * 16 * H];

  // ---- per-feature weights, preloaded into WMMA B-layout registers ----
  v2f w2b[2][8];                // stage-1 B operand: W2[d], [ntile][k-chunk]
  v2f w1v[8], b1v[8];           // per-lane W1/b1 taps for building A in-register
  v2f wml[8];                   // stage-2 B: col0 = Wmu, col1 = Wlv, rest 0
  #pragma unroll
  for (int c = 0; c < 8; ++c) {
    const int k0 = 4 * c + 2 * hi;
    w1v[c] = mkv2(W1[d * H + k0], W1[d * H + k0 + 1]);
    b1v[c] = mkv2(b1[d * H + k0], b1[d * H + k0 + 1]);
    #pragma unroll
    for (int nt = 0; nt < 2; ++nt) {
      const int n = nt * 16 + col;
      w2b[nt][c] = mkv2(W2[d * H * H + k0 * H + n],
                        W2[d * H * H + (k0 + 1) * H + n]);
    }
    const float m0 = Wmu[d * H + k0], m1 = Wmu[d * H + k0 + 1];
    const float l0 = Wlv[d * H + k0], l1 = Wlv[d * H + k0 + 1];
    wml[c] = mkv2(col == 0 ? m0 : (col == 1 ? l0 : 0.0f),
                  col == 0 ? m1 : (col == 1 ? l1 : 0.0f));
  }
  const float b2v0 = b2[d * H + col];
  const float b2v1 = b2[d * H + 16 + col];
  const float bmuv = bmu[d];
  const float blvv = blv[d];

  const int rowBase = blockIdx.x * ROWS_PER_BLOCK + wave * (TILES * 16);

  for (int t = 0; t < TILES; ++t) {
    const int row0 = rowBase + t * 16;
    const float xv = x[(row0 + col) * D + d];   // scalar input to this encoder

    // ---- stage 1: h2 = relu( relu(x*W1+b1) @ W2 + b2 ), 16 rows x 32 cols
    v8f acc0 = {}, acc1 = {};
    #pragma unroll
    for (int c = 0; c < 8; ++c) {
      v2f a;                                     // h1 in A-layout, on the fly
      a.x = fmaxf(fmaf(xv, w1v[c].x, b1v[c].x), 0.0f);
      a.y = fmaxf(fmaf(xv, w1v[c].y, b1v[c].y), 0.0f);
      acc0 = WMMA_F32(a, w2b[0][c], acc0);
      acc1 = WMMA_F32(a, w2b[1][c], acc1);
    }

    // ---- C-layout -> A-layout transpose via private LDS tile
    wave_lds_fence();                            // region reuse vs prev tile
    #pragma unroll
    for (int j = 0; j < 8; ++j) {
      const int r = j + 8 * hi;
      wl[r * H + col]      = fmaxf(acc0[j] + b2v0, 0.0f);
      wl[r * H + 16 + col] = fmaxf(acc1[j] + b2v1, 0.0f);
    }
    wave_lds_fence();

    // ---- stage 2: [mu|lv] = h2 @ [Wmu Wlv 0...], one WMMA chain
    v8f accS = {};
    #pragma unroll
    for (int c = 0; c < 8; ++c) {
      v2f a = *(const v2f*)&wl[col * H + 4 * c + 2 * hi];
      accS = WMMA_F32(a, wml[c], accS);
    }

    // lanes col==0 hold mu, col==1 hold logvar (rows j+8*hi)
    #pragma unroll
    for (int j = 0; j < 8; ++j) {
      const float self  = accS[j];
      const float other = __shfl_xor(self, 1, 32);
      const int row = row0 + j + 8 * hi;
      if (col == 0) {
        const float m  = self + bmuv;
        const float lv = other + blvv;
        mu_out[row * D + d] = m;
        z_ws[row * D + d]   = fmaf(eps[row * D + d], __expf(0.5f * lv), m);
      } else if (col == 1) {
        lv_out[row * D + d] = self + blvv;
      }
    }
  }
}

// ---------------------------------------------------------------------------
// Kernel 2: shared decoder  z[B,64] -> 32 -> 32 -> 1.
// Stage-3 matvec packed into B column 0 so it also runs on the matrix pipe.
// ---------------------------------------------------------------------------
__global__ __launch_bounds__(256) void decoder_kernel(
    const float* __restrict__ z_ws,
    const float* __restrict__ Wd1, const float* __restrict__ bd1,
    const float* __restrict__ Wd2, const float* __restrict__ bd2,
    const float* __restrict__ Wd3, const float* __restrict__ bd3,
    float* __restrict__ action)
{
  __shared__ float lds[WAVES * 16 * H];
  const int lane = threadIdx.x & 31;
  const int wave = threadIdx.x >> 5;
  const int col  = lane & 15;
  const int hi   = lane >> 4;
  float* wl = &lds[wave * 16 * H];

  v2f wd1b[2][16];                   // 64x32 in B-layout, 16 K-chunks
  #pragma unroll
  for (int c = 0; c < 16; ++c) {
    const int k0 = 4 * c + 2 * hi;
    #pragma unroll
    for (int nt = 0; nt < 2; ++nt) {
      const int n = nt * 16 + col;
      wd1b[nt][c] = mkv2(Wd1[k0 * 32 + n], Wd1[(k0 + 1) * 32 + n]);
    }
  }
  v2f wd2b[2][8], wd3b[8];
  #pragma unroll
  for (int c = 0; c < 8; ++c) {
    const int k0 = 4 * c + 2 * hi;
    #pragma unroll
    for (int nt = 0; nt < 2; ++nt) {
      const int n = nt * 16 + col;
      wd2b[nt][c] = mkv2(Wd2[k0 * 32 + n], Wd2[(k0 + 1) * 32 + n]);
    }
    wd3b[c] = mkv2(col == 0 ? Wd3[k0] : 0.0f,
                   col == 0 ? Wd3[k0 + 1] : 0.0f);
  }
  const float bd1v0 = bd1[col], bd1v1 = bd1[16 + col];
  const float bd2v0 = bd2[col], bd2v1 = bd2[16 + col];
  const float bd3v  = bd3[0];

  const int rowBase = blockIdx.x * ROWS_PER_BLOCK + wave * (TILES * 16);

  for (int t = 0; t < TILES; ++t) {
    const int row0 = rowBase + t * 16;

    // ---- a1 = relu(z @ Wd1 + bd1): K=64 -> 16 chunks, A straight from L2
    v8f acc0 = {}, acc1 = {};
    #pragma unroll
    for (int c = 0; c < 16; ++c) {
      v2f a = *(const v2f*)&z_ws[(row0 + col) * 64 + 4 * c + 2 * hi];
      acc0 = WMMA_F32(a, wd1b[0][c], acc0);
      acc1 = WMMA_F32(a, wd1b[1][c], acc1);
    }
    wave_lds_fence();
    #pragma unroll
    for (int j = 0; j < 8; ++j) {
      const int r = j + 8 * hi;
      wl[r * 32 + col]      = fmaxf(acc0[j] + bd1v0, 0.0f);
      wl[r * 32 + 16 + col] = fmaxf(acc1[j] + bd1v1, 0.0f);
    }
    wave_lds_fence();

    // ---- a2 = relu(a1 @ Wd2 + bd2)
    v8f a20 = {}, a21 = {};
    #pragma unroll
    for (int c = 0; c < 8; ++c) {
      v2f a = *(const v2f*)&wl[col * 32 + 4 * c + 2 * hi];
      a20 = WMMA_F32(a, wd2b[0][c], a20);
      a21 = WMMA_F32(a, wd2b[1][c], a21);
    }
    wave_lds_fence();
    #pragma unroll
    for (int j = 0; j < 8; ++j) {
      const int r = j + 8 * hi;
      wl[r * 32 + col]      = fmaxf(a20[j] + bd2v0, 0.0f);
      wl[r * 32 + 16 + col] = fmaxf(a21[j] + bd2v1, 0.0f);
    }
    wave_lds_fence();

    // ---- action = a2 @ Wd3 + bd3 (Wd3 packed into B col 0)
    v8f accS = {};
    #pragma unroll
    for (int c = 0; c < 8; ++c) {
      v2f a = *(const v2f*)&wl[col * 32 + 4 * c + 2 * hi];
      accS = WMMA_F32(a, wd3b[c], accS);
    }
    if (col == 0) {
      #pragma unroll
      for (int j = 0; j < 8; ++j)
        action[row0 + j + 8 * hi] = accS[j] + bd3v;
    }
  }
}

extern "C" void kernel_launch(void* const* d_in, const int* in_sizes, int n_in,
                              void* d_out, int out_size, void* d_ws, size_t ws_size,
                              hipStream_t stream) {
  (void)in_sizes; (void)n_in; (void)out_size; (void)ws_size;
  const float* x   = (const float*)d_in[0];
  const float* eps = (const float*)d_in[1];
  const float* W1  = (const float*)d_in[2];
  const float* b1  = (const float*)d_in[3];
  const float* W2  = (const float*)d_in[4];
  const float* b2  = (const float*)d_in[5];
  const float* Wmu = (const float*)d_in[6];
  const float* bmu = (const float*)d_in[7];
  const float* Wlv = (const float*)d_in[8];
  const float* blv = (const float*)d_in[9];
  const float* Wd1 = (const float*)d_in[10];
  const float* bd1 = (const float*)d_in[11];
  const float* Wd2 = (const float*)d_in[12];
  const float* bd2 = (const float*)d_in[13];
  const float* Wd3 = (const float*)d_in[14];
  const float* bd3 = (const float*)d_in[15];

  float* out     = (float*)d_out;
  float* action  = out;                     // [B]
  float* mu_out  = out + Bsz;               // [B, D]
  float* lv_out  = out + Bsz + Bsz * D;     // [B, D]
  float* z_ws    = (float*)d_ws;            // [B, D] fp32 (8.4 MB, stays in L2)

  dim3 grid1(Bsz / ROWS_PER_BLOCK, D);      // (64, 64)
  encoder_kernel<<<grid1, 256, 0, stream>>>(x, eps, W1, b1, W2, b2,
                                            Wmu, bmu, Wlv, blv,
                                            mu_out, lv_out, z_ws);
  decoder_kernel<<<Bsz / ROWS_PER_BLOCK, 256, 0, stream>>>(z_ws, Wd1, bd1,
                                                           Wd2, bd2, Wd3, bd3,
                                                           action);
}